// MyModel_61933428410125
// MI455X (gfx1250) — compile-verified
//
#include <hip/hip_runtime.h>

// out[t] = sigmoid(in[t]) + carry (carry0 = in[0]) for t=0..7 ; out[8]=out[9]=0.
// in (8,4096,2048) f32 -> out (10,4096,2048) f32.
//
// Memory-bound: 268 MB read + 336 MB write = 604 MB @ 23.3 TB/s -> ~26 us floor.
// Round-1 asm showed ~440 VALU/thread dominated by per-lane 64-bit address
// chains (32 MB slice stride > 24-bit imm offset) plus exp+div sigmoid
// (2 trans/elem). That puts VALU issue near the HBM floor. This version:
//  - passes all 18 slice base pointers as kernargs -> uniform SGPR base +
//    32-bit voffset (saddr-form global_load/store_b128, one v_lshlrev_b32)
//  - col masked to 21 bits so LLVM can prove the byte offset fits 32 bits
//  - sigmoid via v_tanh_f32 (1 trans + 3 VALU/elem), fallback exp2 + v_rcp
//  - ILP=2: second column at +4096 B rides the same address via imm offset
// WMMA inapplicable: elementwise scan, no matrix contraction.

typedef float v4f __attribute__((ext_vector_type(4)));

static constexpr unsigned DIM0   = 8;                        // valid time steps
static constexpr unsigned SLICE4 = 4096u * 2048u / 4u;       // float4s per slice (2 Mi)
static constexpr unsigned ILP    = 2;                        // float4 columns per thread
static constexpr unsigned BLK    = 256;                      // threads per block (8 waves)

struct SlicePtrs {
    const v4f* in[DIM0];   // in  + t*SLICE4
    v4f*       out[10];    // out + t*SLICE4
};

__device__ __forceinline__ float sigmoid1(float x) {
#if __has_builtin(__builtin_amdgcn_tanh_f32)
    return __builtin_fmaf(0.5f, __builtin_amdgcn_tanh_f32(0.5f * x), 0.5f);
#elif __has_builtin(__builtin_amdgcn_tanhf)
    return __builtin_fmaf(0.5f, __builtin_amdgcn_tanhf(0.5f * x), 0.5f);
#else
    // exp2-based: 1/(1+exp(-x)) with raw v_rcp (1-ulp, no Newton step)
    const float e = __builtin_amdgcn_exp2f(x * -1.44269504088896340736f);
    return __builtin_amdgcn_rcpf(1.0f + e);
#endif
}

__device__ __forceinline__ v4f sigmoid4(v4f x) {
    v4f r;
    r.x = sigmoid1(x.x);
    r.y = sigmoid1(x.y);
    r.z = sigmoid1(x.z);
    r.w = sigmoid1(x.w);
    return r;
}

__global__ __launch_bounds__(BLK) void sigscan_kernel(SlicePtrs p) {
    // Mask is a no-op (grid covers exactly SLICE4 columns) but proves to the
    // compiler that col*16 fits in 32 bits -> saddr-form addressing.
    const unsigned col =
        (blockIdx.x * (BLK * ILP) + threadIdx.x) & (SLICE4 - 1u);

    // All 16 loads issued before the dependent carry chain (latency hiding);
    // NT hint: 604 MB streamed once >> 192 MB L2, don't thrash it.
    v4f x[DIM0][ILP];
#pragma unroll
    for (unsigned t = 0; t < DIM0; ++t)
#pragma unroll
        for (unsigned j = 0; j < ILP; ++j)
            x[t][j] = __builtin_nontemporal_load(p.in[t] + col + j * BLK);

    v4f c[ILP];
#pragma unroll
    for (unsigned j = 0; j < ILP; ++j)
        c[j] = x[0][j];

#pragma unroll
    for (unsigned t = 0; t < DIM0; ++t)
#pragma unroll
        for (unsigned j = 0; j < ILP; ++j) {
            c[j] = sigmoid4(x[t][j]) + c[j];            // out[t] -> next carry
            __builtin_nontemporal_store(c[j], p.out[t] + col + j * BLK);
        }

    // Padded steps are exactly zero; d_out is poisoned so write them.
    const v4f z = {0.0f, 0.0f, 0.0f, 0.0f};
#pragma unroll
    for (unsigned t = DIM0; t < 10; ++t)
#pragma unroll
        for (unsigned j = 0; j < ILP; ++j)
            __builtin_nontemporal_store(z, p.out[t] + col + j * BLK);
}

extern "C" void kernel_launch(void* const* d_in, const int* in_sizes, int n_in,
                              void* d_out, int out_size, void* d_ws, size_t ws_size,
                              hipStream_t stream) {
    (void)in_sizes; (void)n_in; (void)d_ws; (void)ws_size; (void)out_size;
    const v4f* in  = (const v4f*)d_in[0];
    v4f*       out = (v4f*)d_out;

    SlicePtrs p;
    for (unsigned t = 0; t < DIM0; ++t) p.in[t]  = in  + (size_t)t * SLICE4;
    for (unsigned t = 0; t < 10;   ++t) p.out[t] = out + (size_t)t * SLICE4;

    const unsigned blocks = SLICE4 / (BLK * ILP);   // 2 Mi / 512 = 4096 blocks
    sigscan_kernel<<<blocks, BLK, 0, stream>>>(p);
}